// TTT_57174604644761
// MI455X (gfx1250) — compile-verified
//
#include <hip/hip_runtime.h>
#include <stdint.h>

// ---------------------------------------------------------------------------
// TTT-style causal linear-attention fusion for MI455X (gfx1250, wave32).
// Out = (2*tril(QK^T) - tril((tril(QK^T)K)K^T)) V, fused per 64-row q-block.
// All GEMMs use V_WMMA_F32_16X16X4_F32 (exact fp32 matrix path).
// Tile loads use GLOBAL_LOAD_ASYNC_TO_LDS_B128 (ASYNCcnt-tracked DMA into
// LDS, no VGPR staging) via inline asm.
// ---------------------------------------------------------------------------

typedef __attribute__((ext_vector_type(2))) float v2f;
typedef __attribute__((ext_vector_type(8))) float v8f;

#define S_LEN 2048
#define H_NUM 16
#define B_NUM 2
#define D_DIM 64
#define BQ    64                 // query rows per workgroup
#define NQB   (S_LEN / BQ)       // 32 query blocks
#define LDSW  68                 // padded LDS row stride (68 % 64 == 4 -> conflict-free)
#define TILE_F (BQ * LDSW)       // floats per 64x64 padded tile

__device__ __forceinline__ v8f wmma4(v2f a, v2f b, v8f c) {
  // (neg_a, A, neg_b, B, c_mod, C, reuse_a, reuse_b)
  return __builtin_amdgcn_wmma_f32_16x16x4_f32(false, a, false, b, (short)0, c,
                                               false, false);
}

__global__ __launch_bounds__(256) void ttt_fused_kernel(
    const float* __restrict__ qkv, float* __restrict__ out) {
  // 3 padded 64x64 fp32 tiles: Q (reused as O2 in phase 2), K/V (shared), S.
  __shared__ float sQ[TILE_F];
  __shared__ float sKV[TILE_F];
  __shared__ float sS[TILE_F];

  const int qb   = blockIdx.x;      // query block
  const int h    = blockIdx.y;      // head
  const int b    = blockIdx.z;      // batch
  const int tid  = threadIdx.x;
  const int wave = tid >> 5;        // 0..7
  const int lane = tid & 31;
  const int ln   = lane & 15;       // lane within half-wave
  const int lh   = lane >> 4;       // half-wave (k-pair / row-group select)
  const int ro   = wave & 3;        // wave's 16-row group (0..3)
  const int cg   = wave >> 2;       // wave's 32-col group (0..1)
  const int qbase = qb * BQ;

  // qkv flat index: (((b*S + s)*3 + which)*H + h)*D + d
  auto grow = [&](int s, int which) -> size_t {
    return ((((size_t)b * S_LEN + s) * 3 + which) * H_NUM + h) * (size_t)D_DIM;
  };

  // Async DMA of one 64x64 fp32 tile into LDS: 16 bytes/lane, 4 reps.
  // VDST VGPR holds the LDS byte offset (low 32 bits of the flat pointer).
  auto loadBlkAsync = [&](float* dst, int sbase, int which) {
#pragma unroll
    for (int rep = 0; rep < 4; ++rep) {
      int idx = rep * 256 + tid;
      int row = idx >> 4;
      int c4  = (idx & 15) << 2;
      const float* gp = qkv + grow(sbase + row, which) + c4;
      unsigned ldsoff = (unsigned)(uintptr_t)(&dst[row * LDSW + c4]);
      asm volatile("global_load_async_to_lds_b128 %0, %1, off"
                   :
                   : "v"(ldsoff), "v"(gp)
                   : "memory");
    }
  };
  auto asyncWait = [&]() {
    asm volatile("s_wait_asynccnt 0x0" ::: "memory");
  };

  // Score tiles: sS = sign * mask(sA * K^T). Wave computes its (ro, cg*32..)
  // 16x32 strip. A-frag & B-frag are both row-major float2 loads (X * Y^T).
  auto scoreTiles = [&](const float* sA, int kbase, bool diag, float sign) {
    v8f s0 = {};
    v8f s1 = {};
#pragma unroll
    for (int k4 = 0; k4 < 16; ++k4) {
      const int col = k4 * 4 + lh * 2;  // k-dim pair for this half-wave
      v2f a  = *(const v2f*)(&sA[(ro * 16 + ln) * LDSW + col]);
      v2f b0 = *(const v2f*)(&sKV[((cg * 2 + 0) * 16 + ln) * LDSW + col]);
      v2f b1 = *(const v2f*)(&sKV[((cg * 2 + 1) * 16 + ln) * LDSW + col]);
      s0 = wmma4(a, b0, s0);
      s1 = wmma4(a, b1, s1);
    }
    // C layout: lanes 0-15 hold rows M=e, lanes 16-31 rows M=e+8, col N=ln.
#pragma unroll
    for (int e = 0; e < 8; ++e) {
      const int row = ro * 16 + e + 8 * lh;
      const int t   = qbase + row;
      const int c0  = cg * 32 + ln;
      const int c1  = cg * 32 + 16 + ln;
      float v0 = s0[e] * sign;
      float v1 = s1[e] * sign;
      if (diag && (kbase + c0) > t) v0 = 0.0f;
      if (diag && (kbase + c1) > t) v1 = 0.0f;
      sS[row * LDSW + c0] = v0;
      sS[row * LDSW + c1] = v1;
    }
  };

  // acc(16x32 strip) += S * B, B = 64x64 tile in sKV (s-major).
  auto gemmSB = [&](v8f& acc0, v8f& acc1) {
#pragma unroll
    for (int k4 = 0; k4 < 16; ++k4) {
      const int col  = k4 * 4 + lh * 2;
      const int srow = col;              // B k-dim rows for this half-wave
      const int c0   = cg * 32 + ln;
      const int c1   = cg * 32 + 16 + ln;
      v2f a  = *(const v2f*)(&sS[(ro * 16 + ln) * LDSW + col]);
      v2f b0 = {sKV[srow * LDSW + c0], sKV[(srow + 1) * LDSW + c0]};
      v2f b1 = {sKV[srow * LDSW + c1], sKV[(srow + 1) * LDSW + c1]};
      acc0 = wmma4(a, b0, acc0);
      acc1 = wmma4(a, b1, acc1);
    }
  };

  v8f o2a = {}, o2b = {};    // O2 strip accumulators (16x32 per wave)
  v8f outa = {}, outb = {};  // output strip accumulators

  // Load Q block once (async DMA; published by the first loop barrier).
  loadBlkAsync(sQ, qbase, 0);
  asyncWait();

  // ---- Phase 1: O2 += tril(QK^T)K ; P1 += tril(QK^T)V -------------------
  for (int kb = 0; kb <= qb; ++kb) {
    const int kbase = kb * BQ;
    __syncthreads();                       // prev iter readers of sKV/sS done
    loadBlkAsync(sKV, kbase, 1);           // K
    asyncWait();
    __syncthreads();
    scoreTiles(sQ, kbase, kb == qb, 1.0f); // sS = tril(Q K^T) block
    __syncthreads();
    gemmSB(o2a, o2b);                      // O2 += S * K
    __syncthreads();                       // K readers done
    loadBlkAsync(sKV, kbase, 2);           // V (overwrites K)
    asyncWait();
    __syncthreads();
    gemmSB(outa, outb);                    // P1 += S * V
  }

  // ---- Transition: Out = 2*P1; park O2 in sQ (Q is dead) ----------------
  __syncthreads();
#pragma unroll
  for (int e = 0; e < 8; ++e) {
    const int row = ro * 16 + e + 8 * lh;
    sQ[row * LDSW + cg * 32 + ln]      = o2a[e];
    sQ[row * LDSW + cg * 32 + 16 + ln] = o2b[e];
  }
  outa *= 2.0f;
  outb *= 2.0f;

  // ---- Phase 2: Out -= tril(O2 K^T) V -----------------------------------
  for (int kb = 0; kb <= qb; ++kb) {
    const int kbase = kb * BQ;
    __syncthreads();                        // O2 visible / prev readers done
    loadBlkAsync(sKV, kbase, 1);            // K
    asyncWait();
    __syncthreads();
    scoreTiles(sQ, kbase, kb == qb, -1.0f); // sS = -tril(O2 K^T) block
    __syncthreads();
    loadBlkAsync(sKV, kbase, 2);            // V
    asyncWait();
    __syncthreads();
    gemmSB(outa, outb);                     // Out += (-A2) * V
  }

  // ---- Writeback: out[b, t, h, d] ---------------------------------------
#pragma unroll
  for (int e = 0; e < 8; ++e) {
    const int t = qbase + ro * 16 + e + 8 * lh;
    const size_t g = (((size_t)b * S_LEN + t) * H_NUM + h) * (size_t)D_DIM;
    out[g + cg * 32 + ln]      = outa[e];
    out[g + cg * 32 + 16 + ln] = outb[e];
  }
}

extern "C" void kernel_launch(void* const* d_in, const int* in_sizes, int n_in,
                              void* d_out, int out_size, void* d_ws,
                              size_t ws_size, hipStream_t stream) {
  (void)in_sizes; (void)n_in; (void)d_ws; (void)ws_size; (void)out_size;
  const float* qkv = (const float*)d_in[0];
  float* out = (float*)d_out;
  dim3 grid(NQB, H_NUM, B_NUM);  // 32 x 16 x 2 = 1024 workgroups
  dim3 block(256);               // 8 wave32 waves
  ttt_fused_kernel<<<grid, block, 0, stream>>>(qkv, out);
}